// SourceGenerator_76519137345763
// MI455X (gfx1250) — compile-verified
//
#include <hip/hip_runtime.h>

typedef __attribute__((ext_vector_type(2))) float v2f;
typedef __attribute__((ext_vector_type(8))) float v8f;

#ifndef LN_EPS
#define LN_EPS 1e-3f
#endif

// ---------------------------------------------------------------------------
// Kernel 1: per-batch column mean over N rows + gather of F=2 indexed rows.
// One block (256 threads) per batch element. Streams H once (bandwidth bound).
// emb layout per batch b: [H[b,i0,:] (128) | H[b,i1,:] (128) | mean (128)]
// ---------------------------------------------------------------------------
__global__ __launch_bounds__(256) void gather_mean_kernel(
    const float4* __restrict__ H4,     // (B, N, 32) float4 view of (B,N,128)
    const int*    __restrict__ indice, // (B, 2)
    float*        __restrict__ emb,    // (B, 384)
    int N)
{
    const int b  = blockIdx.x;
    const int t  = threadIdx.x;
    const int c4 = t & 31;   // which float4 column (0..31 -> 128 floats)
    const int rg = t >> 5;   // row group 0..7

    const float4* Hb = H4 + (size_t)b * N * 32;

    float4 acc = make_float4(0.f, 0.f, 0.f, 0.f);
    #pragma unroll 5
    for (int n = rg; n < N; n += 8) {
        float4 v = Hb[(size_t)n * 32 + c4];
        acc.x += v.x; acc.y += v.y; acc.z += v.z; acc.w += v.w;
    }

    __shared__ float4 part[8][32];
    part[rg][c4] = acc;
    __syncthreads();

    if (t < 32) {
        float4 s = part[0][t];
        #pragma unroll
        for (int r = 1; r < 8; ++r) {
            float4 v = part[r][t];
            s.x += v.x; s.y += v.y; s.z += v.z; s.w += v.w;
        }
        const float inv = 1.0f / (float)N;
        float4 m = make_float4(s.x * inv, s.y * inv, s.z * inv, s.w * inv);
        ((float4*)(emb + (size_t)b * 384 + 256))[t] = m;
    } else if (t < 96) {
        const int f = (t - 32) >> 5;   // 0 or 1
        const int c = (t - 32) & 31;
        const int idx = indice[b * 2 + f];
        float4 v = Hb[(size_t)idx * 32 + c];
        ((float4*)(emb + (size_t)b * 384 + f * 128))[c] = v;
    }
}

// ---------------------------------------------------------------------------
// Kernel 2: x = relu(emb @ W + b); LayerNorm over D=128; uses
// V_WMMA_F32_16X16X4_F32. Block = 256 threads (8 waves); block owns a 16-row
// M-tile of emb; wave w owns columns [16w, 16w+16). K = 384 -> 96 WMMA steps.
//
// ISA VGPR layouts (CDNA5 ISA):
//   A 16x4 f32 : lanes 0-15 -> {K=k+0, K=k+1}, lanes 16-31 -> {K=k+2, K=k+3}
//   B 4x16 f32 : lanes 0-15 -> {K=k+0, K=k+1} at N=lane, lanes 16-31 -> {K=k+2,k+3}
//   C/D 16x16  : VGPR r, lanes 0-15 -> M=r, lanes 16-31 -> M=r+8
// ---------------------------------------------------------------------------
__global__ __launch_bounds__(256) void gemm_relu_ln_kernel(
    const float* __restrict__ emb,    // (B, 384)
    const float* __restrict__ W,      // (384, 128)
    const float* __restrict__ bias,   // (128,)
    const float* __restrict__ gamma,  // (128,)
    const float* __restrict__ beta,   // (128,)
    float*       __restrict__ out)    // (B, 1, 128)
{
    constexpr int K = 384;
    constexpr int D = 128;

    const int row_base = blockIdx.x * 16;
    const int wave = threadIdx.x >> 5;       // 0..7 -> column tile
    const int lane = threadIdx.x & 31;
    const int half = lane >> 4;              // 0: lanes 0-15, 1: lanes 16-31
    const int li   = lane & 15;
    const int col_base = wave * 16;

    const float* __restrict__ Arow = emb + (size_t)(row_base + li) * K;

    v8f acc = {};
    for (int k = 0; k < K; k += 4) {
        const int ka = k + 2 * half;
        v2f a, bfrag;
        a.x     = Arow[ka];
        a.y     = Arow[ka + 1];
        bfrag.x = W[(size_t)ka * D + col_base + li];
        bfrag.y = W[(size_t)(ka + 1) * D + col_base + li];
        acc = __builtin_amdgcn_wmma_f32_16x16x4_f32(
            /*neg_a=*/false, a, /*neg_b=*/false, bfrag,
            /*c_mod=*/(short)0, acc, /*reuse_a=*/false, /*reuse_b=*/false);
    }

    // Scatter accumulator tile -> LDS with bias + ReLU applied.
    __shared__ float tile[16][132];          // stride 132 avoids bank conflicts
    __shared__ float rmean[16];
    __shared__ float rrstd[16];

    {
        const float bc = bias[col_base + li];
        const int r0 = half * 8;
        #pragma unroll
        for (int vr = 0; vr < 8; ++vr) {
            float v = acc[vr] + bc;
            tile[r0 + vr][col_base + li] = fmaxf(v, 0.0f);
        }
    }
    __syncthreads();

    // Per-row mean / biased variance (matches jnp.var) over D=128.
    if (threadIdx.x < 16) {
        const int r = threadIdx.x;
        float s = 0.f, s2 = 0.f;
        #pragma unroll 8
        for (int c = 0; c < D; ++c) {
            float v = tile[r][c];
            s += v; s2 += v * v;
        }
        const float mean = s * (1.0f / D);
        const float var  = s2 * (1.0f / D) - mean * mean;
        rmean[r] = mean;
        rrstd[r] = rsqrtf(var + LN_EPS);
    }
    __syncthreads();

    // Normalize + affine + store: 16 rows x 128 cols = 2048 elems / 256 thr.
    for (int i = threadIdx.x; i < 16 * D; i += 256) {
        const int r = i >> 7;
        const int c = i & (D - 1);
        float v = (tile[r][c] - rmean[r]) * rrstd[r] * gamma[c] + beta[c];
        out[(size_t)(row_base + r) * D + c] = v;
    }
}

// ---------------------------------------------------------------------------
// Host entry point.
// Inputs (setup_inputs order): H (B,N,128) f32, indice (B,2) i32,
// W (384,128) f32, b (128) f32, gamma (128) f32, beta (128) f32.
// Output: (B,1,128) f32.
// ---------------------------------------------------------------------------
extern "C" void kernel_launch(void* const* d_in, const int* in_sizes, int n_in,
                              void* d_out, int out_size, void* d_ws, size_t ws_size,
                              hipStream_t stream) {
    const float* H      = (const float*)d_in[0];
    const int*   indice = (const int*)  d_in[1];
    const float* W      = (const float*)d_in[2];
    const float* bias   = (const float*)d_in[3];
    const float* gamma  = (const float*)d_in[4];
    const float* beta   = (const float*)d_in[5];
    float* out = (float*)d_out;

    const int D = 128;
    const int B = in_sizes[1] / 2;                 // indice is (B, 2)
    const int N = in_sizes[0] / (B * D);           // H is (B, N, D)

    float* emb = (float*)d_ws;                     // (B, 384) scratch

    gather_mean_kernel<<<B, 256, 0, stream>>>(
        (const float4*)H, indice, emb, N);

    gemm_relu_ln_kernel<<<B / 16, 256, 0, stream>>>(
        emb, W, bias, gamma, beta, out);
}